// net_4698694222647
// MI455X (gfx1250) — compile-verified
//
#include <hip/hip_runtime.h>

#define N_NODES 50000
#define N_EDGES 800000
#define IN_CH   128
#define HID_CH  512
#define OUT_CH  256

#define MT        5                 // M tiles (16 rows each) per block
#define M_PER_BLK (MT * 16)         // 80 rows per block; 50000 / 80 = 625
#define PADW      132               // padded LDS row stride (floats) -> no bank conflicts

typedef float    v2f  __attribute__((ext_vector_type(2)));
typedef float    v8f  __attribute__((ext_vector_type(8)));

// ---------------------------------------------------------------------------
// Kernel 1: h = x  (residual init; scatter then accumulates on top)
// ---------------------------------------------------------------------------
__global__ void __launch_bounds__(256)
copy_f4_kernel(const float4* __restrict__ src, float4* __restrict__ dst, int n4) {
    int i = blockIdx.x * blockDim.x + threadIdx.x;
    if (i < n4) dst[i] = src[i];
}

// ---------------------------------------------------------------------------
// Kernel 2: h[dst] += x[src] for each edge. One wave32 per edge; each lane
// owns 4 contiguous channels (float4 load, 4 hardware f32 atomics).
// ---------------------------------------------------------------------------
__global__ void __launch_bounds__(256)
scatter_edges_kernel(const float* __restrict__ x, const int* __restrict__ ei,
                     float* __restrict__ h) {
    const int wave = threadIdx.x >> 5;
    const int lane = threadIdx.x & 31;
    const int e    = blockIdx.x * 8 + wave;      // grid sized so e < N_EDGES
    const int src  = ei[e];                      // row 0 of edge_index
    const int dst  = ei[N_EDGES + e];            // row 1 of edge_index
    const float4 v = *(const float4*)(x + (size_t)src * IN_CH + lane * 4);
    float* p = h + (size_t)dst * IN_CH + lane * 4;
    unsafeAtomicAdd(p + 0, v.x);
    unsafeAtomicAdd(p + 1, v.y);
    unsafeAtomicAdd(p + 2, v.z);
    unsafeAtomicAdd(p + 3, v.w);
}

// ---------------------------------------------------------------------------
// Kernel 3: Wc = W1 @ W2   [IN_CH x OUT_CH], bc = b1 @ W2  [OUT_CH]
// (h@W1+b1)@W2 == h@(W1@W2) + b1@W2 — no nonlinearity between the linears.
// ---------------------------------------------------------------------------
__global__ void __launch_bounds__(256)
fuse_weights_kernel(const float* __restrict__ W1, const float* __restrict__ b1,
                    const float* __restrict__ W2, float* __restrict__ Wc,
                    float* __restrict__ bc) {
    const int j = threadIdx.x;                   // 0..255 = output column
    if ((int)blockIdx.x < IN_CH) {
        const int i = blockIdx.x;
        float acc = 0.0f;
        for (int k = 0; k < HID_CH; ++k)
            acc = fmaf(W1[i * HID_CH + k], W2[k * OUT_CH + j], acc);
        Wc[i * OUT_CH + j] = acc;
    } else {
        float acc = 0.0f;
        for (int k = 0; k < HID_CH; ++k)
            acc = fmaf(b1[k], W2[k * OUT_CH + j], acc);
        bc[j] = acc;
    }
}

// ---------------------------------------------------------------------------
// Kernel 4: out = h @ Wc + bc.
// 512 threads = 16 waves; wave w owns N-tile w with its 128x16 B fragment
// register-resident (64 VGPRs), and sweeps 5 M-tiles. The 80x128 A slab is
// staged once per block into LDS via async global->LDS b128 copies
// (ASYNCcnt path), padded to 132 floats/row for conflict-free ds_load_b64.
// ---------------------------------------------------------------------------
__global__ void __launch_bounds__(512)
gemm_wmma_kernel(const float* __restrict__ H, const float* __restrict__ Wc,
                 const float* __restrict__ bc, float* __restrict__ out) {
    __shared__ float As[M_PER_BLK * PADW];       // 80*132*4 = 42240 B

    const int tid   = threadIdx.x;
    const int lane  = tid & 31;
    const int wave  = tid >> 5;                  // 0..15 = N tile index
    const int mBase = blockIdx.x * M_PER_BLK;
    const int nBase = wave * 16;
    const int hsel  = lane >> 4;                 // 16-lane half select
    const int lidx  = lane & 15;

    // ---- stage A (80 x 128 f32) into LDS: 2560 float4 chunks, 5/thread ----
    #pragma unroll
    for (int i = 0; i < 5; ++i) {
        const int chunk = tid + i * 512;         // 0..2559
        const int row   = chunk >> 5;            // 32 float4 chunks per row
        const int col4  = chunk & 31;
        const float* gp = H + (size_t)(mBase + row) * IN_CH + col4 * 4;
        const unsigned lp =
            (unsigned)(uintptr_t)(const void*)&As[row * PADW + col4 * 4];
        asm volatile("global_load_async_to_lds_b128 %0, %1, off"
                     :: "v"(lp), "v"((unsigned long long)(uintptr_t)gp)
                     : "memory");
    }

    // ---- preload B fragments for this wave's N tile (overlap async A) ----
    v2f bfrag[32];
    {
        const float* wCol = Wc + nBase + lidx + (size_t)(hsel * 2) * OUT_CH;
        #pragma unroll
        for (int kk = 0; kk < 32; ++kk) {
            bfrag[kk].x = wCol[(size_t)(kk * 4 + 0) * OUT_CH];
            bfrag[kk].y = wCol[(size_t)(kk * 4 + 1) * OUT_CH];
        }
    }
    const float bias = bc[nBase + lidx];

    asm volatile("s_wait_asynccnt 0x0" ::: "memory");
    __syncthreads();

    // ---- 5 M tiles; A from LDS, B from registers, 32 WMMAs per tile ----
    for (int mt = 0; mt < MT; ++mt) {
        const float* aRow = &As[(mt * 16 + lidx) * PADW + hsel * 2];
        v8f acc = {};
        #pragma unroll
        for (int kk = 0; kk < 32; ++kk) {
            v2f a;
            a.x = aRow[kk * 4 + 0];
            a.y = aRow[kk * 4 + 1];
            acc = __builtin_amdgcn_wmma_f32_16x16x4_f32(
                false, a, false, bfrag[kk], (short)0, acc, false, false);
        }
        const int rowB = mBase + mt * 16 + hsel * 8;
        const int col  = nBase + lidx;
        #pragma unroll
        for (int r = 0; r < 8; ++r)
            out[(size_t)(rowB + r) * OUT_CH + col] = acc[r] + bias;
    }
}

// ---------------------------------------------------------------------------
extern "C" void kernel_launch(void* const* d_in, const int* in_sizes, int n_in,
                              void* d_out, int out_size, void* d_ws, size_t ws_size,
                              hipStream_t stream) {
    const float* x  = (const float*)d_in[0];   // [N_NODES, IN_CH]
    const int*   ei = (const int*)  d_in[1];   // [2, N_EDGES]
    const float* W1 = (const float*)d_in[2];   // [IN_CH, HID_CH]
    const float* b1 = (const float*)d_in[3];   // [HID_CH]
    const float* W2 = (const float*)d_in[4];   // [HID_CH, OUT_CH]
    float* out = (float*)d_out;                // [N_NODES, OUT_CH]

    // Workspace layout: h | Wc | bc   (~25.8 MB total)
    float* h  = (float*)d_ws;
    float* Wc = h  + (size_t)N_NODES * IN_CH;
    float* bc = Wc + (size_t)IN_CH * OUT_CH;

    // 1) h = x
    {
        const int n4 = N_NODES * IN_CH / 4;
        copy_f4_kernel<<<(n4 + 255) / 256, 256, 0, stream>>>(
            (const float4*)x, (float4*)h, n4);
    }
    // 2) h[dst] += x[src]  (8 edges per 256-thread block; 800000/8 = 100000)
    scatter_edges_kernel<<<N_EDGES / 8, 256, 0, stream>>>(x, ei, h);
    // 3) Wc = W1@W2, bc = b1@W2
    fuse_weights_kernel<<<IN_CH + 1, 256, 0, stream>>>(W1, b1, W2, Wc, bc);
    // 4) out = h@Wc + bc   (625 blocks x 80 rows, 16 waves = 16 N tiles)
    gemm_wmma_kernel<<<N_NODES / M_PER_BLK, 512, 0, stream>>>(h, Wc, bc, out);
}